// SCNN_81372450390478
// MI455X (gfx1250) — compile-verified
//
#include <hip/hip_runtime.h>
#include <hip/hip_bf16.h>

// SCNN (4 directional sequential scans, 9-tap conv along channels) for
// MI455X / gfx1250. Conv-along-C is a banded 128x128 matmul -> chained
// V_WMMA_F32_16X16X4_F32 (6 WMMAs per 16ch x 16col tile per step, split
// into two independent 3-deep chains to halve the WMMA RAW critical path).
// xi global loads are software-pipelined one step ahead so their latency
// overlaps the WMMA chain + barrier instead of serializing each step.
//
// Layout trace of the reference (flips are never undone):
//   pass D: reads [n][c][h][w]      row s,      writes [n][c][h][w]     row s
//   pass U: reads [n][c][h][w]      row 159-s,  writes [n][c][w][hf]    col s (transposed)
//   pass L: reads [n][c][w][hf]     row s,      writes [n][c][w][hf]    row s
//   pass R: reads [n][c][w][hf]     row 159-s,  writes out[n][c][hf][wf] col s (transposed)
// => D,L use rev=0 ; U,R use rev=1. Ping-pong: x->ws->out->ws->out.

#define KW   9
#define SDIM 160
#define CDIM 128
#define COLP 20          // LDS column stride (banks: 8*20 % 64 = 32, 2*20 % 64 = 40 -> conflict-free)

typedef __attribute__((ext_vector_type(2))) float v2f;
typedef __attribute__((ext_vector_type(8))) float v8f;

__device__ __forceinline__ v8f wmma_f32_k4(v2f a, v2f b, v8f c) {
  // D(16x16,f32) = A(16x4,f32) x B(4x16,f32) + C
  return __builtin_amdgcn_wmma_f32_16x16x4_f32(false, a, false, b, (short)0, c,
                                               false, false);
}

__global__ __launch_bounds__(256) void scnn_pass(const float* __restrict__ in,
                                                 float* __restrict__ out,
                                                 const float* __restrict__ wv,
                                                 const float* __restrict__ bv,
                                                 int rev) {
  // recurrence state (non-relu'd), double buffered: [buf][channel][col]
  __shared__ float prev[2][CDIM][COLP];
  __shared__ float wsh[16];

  const int tid  = threadIdx.x;
  const int lane = tid & 31;
  const int lo   = lane & 15;        // column / N index, and A-matrix row m
  const int hi   = lane >> 4;        // half-wave select
  const int I    = __builtin_amdgcn_readfirstlane(tid >> 5);  // channel tile 0..7

  if (tid < KW) wsh[tid] = wv[tid];
  __syncthreads();

  const float bias = bv[0];

  // ---- Build A operands (band matrix tiles), ISA 16x4 A layout:
  //      m = lane&15, k_local = v + 2*hi.
  // central block (J==I, kb=0,4,8,12): w[kb + k_local - m + 4]
  // J=I-1 sliver (kb=12):              w[kb + k_local - m + 4 - 16]
  // J=I+1 sliver (kb=0):               w[kb + k_local - m + 4 + 16]
  auto band = [&](int i) -> float { return (i >= 0 && i < KW) ? wsh[i] : 0.0f; };
  v2f A_m1, A_p1, A_c0, A_c1, A_c2, A_c3;
#pragma unroll
  for (int v = 0; v < 2; ++v) {
    const int kl = v + 2 * hi;
    A_c0[v] = band(0  + kl - lo + 4);
    A_c1[v] = band(4  + kl - lo + 4);
    A_c2[v] = band(8  + kl - lo + 4);
    A_c3[v] = band(12 + kl - lo + 4);
    // zero halo A for edge tiles -> branchless WMMA with full EXEC
    A_m1[v] = (I > 0) ? band(12 + kl - lo + 4 - 16) : 0.0f;
    A_p1[v] = (I < 7) ? band(0  + kl - lo + 4 + 16) : 0.0f;
  }

  // ---- Column group: 1280 columns = 8 n * 160 pos, 16 per workgroup
  const int g  = blockIdx.x;        // 0..79
  const int n  = g / 10;
  const int q  = (g % 10) * 16 + lo;

  // D/C layout: VGPR r <-> channel 16I + r + 8*hi, column q
  const size_t cb = (size_t)(n * CDIM + I * 16 + hi * 8) * (size_t)(SDIM * SDIM);

  // B-operand LDS channel bases (clamped; edge values multiplied by zero A)
  const int ch0 = I * 16 + 2 * hi;
  const int chm = (ch0 >= 4) ? (ch0 - 4) : 0;
  const int chp = (ch0 + 16 <= CDIM - 2) ? (ch0 + 16) : (CDIM - 2);

  // ---- Step 0: out[0] = x[0] (no conv); keep non-relu value as state.
  // Also pre-issue the xi loads for step 1 (software pipeline stage 0).
  float xi[8];
  {
    const int a0 = rev ? (SDIM - 1) : 0;
    const int a1 = rev ? (SDIM - 2) : 1;
    float x0[8];
#pragma unroll
    for (int r = 0; r < 8; ++r)
      x0[r] = in[cb + (size_t)r * (SDIM * SDIM) + (size_t)a0 * SDIM + q];
#pragma unroll
    for (int r = 0; r < 8; ++r)
      xi[r] = in[cb + (size_t)r * (SDIM * SDIM) + (size_t)a1 * SDIM + q];
#pragma unroll
    for (int r = 0; r < 8; ++r) {
      prev[0][I * 16 + r + hi * 8][lo] = x0[r];
      const size_t pr = cb + (size_t)r * (SDIM * SDIM);
      const size_t oa = rev ? (pr + (size_t)q * SDIM) : (pr + q);
      out[oa] = fmaxf(x0[r], 0.0f);
    }
  }
  __syncthreads();

  // ---- Sequential scan
  for (int s = 1; s < SDIM; ++s) {
    const int pb = (s - 1) & 1;
    const int nb = s & 1;

    // B operands (4x16, assumed layout n=lane&15, k = v + 2*hi)
    v2f Bc0, Bc1, Bc2, Bc3, Bm1, Bp1;
#pragma unroll
    for (int v = 0; v < 2; ++v) {
      Bc0[v] = prev[pb][ch0 + 0  + v][lo];
      Bc1[v] = prev[pb][ch0 + 4  + v][lo];
      Bc2[v] = prev[pb][ch0 + 8  + v][lo];
      Bc3[v] = prev[pb][ch0 + 12 + v][lo];
      Bm1[v] = prev[pb][chm + v][lo];
      Bp1[v] = prev[pb][chp + v][lo];
    }

    // Pipeline: issue next step's xi loads now; their wait lands after the
    // WMMA chain. Address clamped on the last iteration (result unused).
    const int an = rev ? ((s + 1 < SDIM) ? (SDIM - 2 - s) : 0)
                       : ((s + 1 < SDIM) ? (s + 1) : (SDIM - 1));
    float xin[8];
#pragma unroll
    for (int r = 0; r < 8; ++r)
      xin[r] = in[cb + (size_t)r * (SDIM * SDIM) + (size_t)an * SDIM + q];

    // Two independent 3-deep WMMA chains (halves D->C RAW latency chain)
    v8f acc0, acc1;
#pragma unroll
    for (int e = 0; e < 8; ++e) { acc0[e] = bias; acc1[e] = 0.0f; }

    acc0 = wmma_f32_k4(A_m1, Bm1, acc0);   // left halo (zero-A at I==0)
    acc1 = wmma_f32_k4(A_c2, Bc2, acc1);
    acc0 = wmma_f32_k4(A_c0, Bc0, acc0);
    acc1 = wmma_f32_k4(A_c3, Bc3, acc1);
    acc0 = wmma_f32_k4(A_c1, Bc1, acc0);
    acc1 = wmma_f32_k4(A_p1, Bp1, acc1);   // right halo (zero-A at I==7)

#pragma unroll
    for (int r = 0; r < 8; ++r) {
      const float cv = fmaxf(acc0[r] + acc1[r], 0.0f) + xi[r]; // scan state
      prev[nb][I * 16 + r + hi * 8][lo] = cv;
      const size_t pr = cb + (size_t)r * (SDIM * SDIM);
      const size_t oa = rev ? (pr + (size_t)q * SDIM + s)
                            : (pr + (size_t)s * SDIM + q);
      out[oa] = fmaxf(cv, 0.0f);                               // outer relu
    }

#pragma unroll
    for (int r = 0; r < 8; ++r) xi[r] = xin[r];

    // prefetch two steps ahead (gfx1250 global_prefetch_b8)
    if (s + 2 < SDIM) {
      const int ap = rev ? (SDIM - 3 - s) : (s + 2);
      __builtin_prefetch(&in[cb + (size_t)ap * SDIM + q], 0, 3);
    }
    __syncthreads();
  }
}

extern "C" void kernel_launch(void* const* d_in, const int* in_sizes, int n_in,
                              void* d_out, int out_size, void* d_ws, size_t ws_size,
                              hipStream_t stream) {
  (void)in_sizes; (void)n_in; (void)out_size; (void)ws_size;
  const float* x   = (const float*)d_in[0];
  const float* w_d = (const float*)d_in[1];
  const float* b_d = (const float*)d_in[2];
  const float* w_u = (const float*)d_in[3];
  const float* b_u = (const float*)d_in[4];
  const float* w_l = (const float*)d_in[5];
  const float* b_l = (const float*)d_in[6];
  const float* w_r = (const float*)d_in[7];
  const float* b_r = (const float*)d_in[8];
  float* outp = (float*)d_out;
  float* tmp  = (float*)d_ws;   // needs 8*128*160*160*4 = 100 MB scratch

  dim3 grid(80), block(256);
  // down:  x   -> tmp   (row-major in/out)
  scnn_pass<<<grid, block, 0, stream>>>(x,    tmp,  w_d, b_d, 0);
  // up:    tmp -> out   (reversed read, transposed write -> [n][c][w][hf])
  scnn_pass<<<grid, block, 0, stream>>>(tmp,  outp, w_u, b_u, 1);
  // left:  out -> tmp
  scnn_pass<<<grid, block, 0, stream>>>(outp, tmp,  w_l, b_l, 0);
  // right: tmp -> out   (final layout [n][c][hf][wf], matching reference)
  scnn_pass<<<grid, block, 0, stream>>>(tmp,  outp, w_r, b_r, 1);
}